// SpMM_16338055594698
// MI455X (gfx1250) — compile-verified
//
#include <hip/hip_runtime.h>

// SpMM scatter (cogdl spmm fallback math) for MI455X / gfx1250.
//
// out = in_norm * segment_sum( (out_norm*x)[col] * values, row )
//
// Fused single scatter pass: per edge e,
//   out[row[e], :] += (values[e]*out_norm[col[e]]*in_norm[row[e]]) * x[col[e], :]
// (in_norm[row] is constant per destination row, so folding it into the edge
//  weight is exact w.r.t. the reference's final elementwise multiply.)
//
// Mapping: one wave32 per edge iteration. Lane l handles floats [4l,4l+4) and
// [128+4l,128+4l+4) of the 256-float row: two coalesced global_load_b128 per
// lane + 8 non-returning global_atomic_add_f32 into the (L2-resident, 102MB
// < 192MB L2) output. Edge scalars go through readfirstlane -> SMEM s_load.
// Next edge's source row is software-prefetched (global_prefetch_b8).
//
// NOTE: scatter kernel is first in the file so the disasm snippet shows the
// atomic/prefetch path.

#define D_FEAT 256

__global__ __launch_bounds__(256) void spmm_scatter_kernel(
    const float* __restrict__ x,
    const int* __restrict__ row,
    const int* __restrict__ col,
    const float* __restrict__ values,
    const float* __restrict__ out_norm,
    const float* __restrict__ in_norm,
    float* __restrict__ out,
    int num_edges) {
  const int lane = threadIdx.x & 31;
  const int wave = blockIdx.x * (blockDim.x >> 5) + (threadIdx.x >> 5);
  const int total_waves = gridDim.x * (blockDim.x >> 5);

  for (int e0 = wave; e0 < num_edges; e0 += total_waves) {
    // Wave-uniform edge index -> scalar (SGPR) so edge metadata uses s_load.
    const int e = __builtin_amdgcn_readfirstlane(e0);

    // Prefetch next iteration's source row (gfx1250 global_prefetch_b8).
    // 16 lanes x 64B cover the 1KB row.
    const int en = e + total_waves;
    if (en < num_edges) {
      const int cn = __builtin_amdgcn_readfirstlane(col[en]);
      __builtin_prefetch(x + (size_t)cn * D_FEAT + (size_t)(lane & 15) * 16, 0, 3);
    }

    const int r = __builtin_amdgcn_readfirstlane(row[e]);
    const int c = __builtin_amdgcn_readfirstlane(col[e]);
    const float w = values[e] * out_norm[c] * in_norm[r];

    const float4* __restrict__ xr =
        (const float4*)(x + (size_t)c * D_FEAT);
    const float4 a = xr[lane];        // floats [4*lane, 4*lane+4)
    const float4 b = xr[lane + 32];   // floats [128+4*lane, ...)

    float* o = out + (size_t)r * D_FEAT + 4 * lane;
    unsafeAtomicAdd(o + 0,   w * a.x);
    unsafeAtomicAdd(o + 1,   w * a.y);
    unsafeAtomicAdd(o + 2,   w * a.z);
    unsafeAtomicAdd(o + 3,   w * a.w);
    unsafeAtomicAdd(o + 128, w * b.x);
    unsafeAtomicAdd(o + 129, w * b.y);
    unsafeAtomicAdd(o + 130, w * b.z);
    unsafeAtomicAdd(o + 131, w * b.w);
  }
}

__global__ __launch_bounds__(256) void spmm_zero_kernel(float4* __restrict__ out,
                                                        int n4) {
  int i = blockIdx.x * blockDim.x + threadIdx.x;
  const int stride = gridDim.x * blockDim.x;
  const float4 z = make_float4(0.f, 0.f, 0.f, 0.f);
  for (; i < n4; i += stride) out[i] = z;
}

extern "C" void kernel_launch(void* const* d_in, const int* in_sizes, int n_in,
                              void* d_out, int out_size, void* d_ws, size_t ws_size,
                              hipStream_t stream) {
  const float* x        = (const float*)d_in[0];
  const int*   row      = (const int*)d_in[1];
  const int*   col      = (const int*)d_in[2];
  const float* values   = (const float*)d_in[3];
  const float* out_norm = (const float*)d_in[4];
  const float* in_norm  = (const float*)d_in[5];
  float* out = (float*)d_out;

  const int E  = in_sizes[1];     // edge count (row array length)
  const int n4 = out_size / 4;    // N*D/4 float4s

  // 1) Zero the accumulator (must run every call: scatter accumulates and the
  //    harness replays without re-poisoning d_out).
  spmm_zero_kernel<<<4096, 256, 0, stream>>>((float4*)out, n4);

  // 2) Fused gather-scale-scatter. 40960 blocks x 8 waves = 327680 waves,
  //    ~10 edges per wave (grid-stride keeps consecutive waves on consecutive
  //    edges for edge-array locality + enables next-row prefetch).
  spmm_scatter_kernel<<<40960, 256, 0, stream>>>(x, row, col, values,
                                                 out_norm, in_norm, out, E);
}